// Norm_730144440413
// MI455X (gfx1250) — compile-verified
//
#include <hip/hip_runtime.h>

typedef __attribute__((ext_vector_type(2))) float v2f;
typedef __attribute__((ext_vector_type(8))) float v8f;

#define CLN_EPS 1e-8f

// ---------------------------------------------------------------------------
// Phase 1: per (b, 16-wide l tile) compute s[l] = sum_m x[b,m,l] and
// s2[l] = sum_m x^2 using V_WMMA_F32_16X16X4_F32 with an all-ones A matrix.
// D[i,j] = sum_k 1 * B[k,j]; accumulating over m-chunks of 4 gives the full
// column sum in every row of the 16x16 f32 accumulator. Since A is all ones,
// the result does not depend on which K slot each m-row occupies — only on
// the N = lane%16 striping of the B operand.
// One wave (32 lanes) per tile. EXEC all-ones inside the wave (WMMA req).
// ---------------------------------------------------------------------------
__global__ void __launch_bounds__(256) cln_colsum_wmma(
    const float* __restrict__ x, float* __restrict__ s, float* __restrict__ s2,
    int B, int M, int L)
{
  const int lane = (int)(threadIdx.x & 31);
  const int wave = (int)((blockIdx.x * blockDim.x + threadIdx.x) >> 5);
  const int tilesPerB = L >> 4;                 // 16 l-values per tile
  if (wave >= B * tilesPerB) return;            // wave-uniform guard
  const int b  = wave / tilesPerB;
  const int l0 = (wave % tilesPerB) << 4;

  // lanes 0..15 stream row m, lanes 16..31 stream row m+1 (consecutive l's:
  // two coalesced 64B segments per load).
  const float* px = x + (size_t)b * (size_t)M * (size_t)L
                      + (lane >= 16 ? (size_t)L : (size_t)0)
                      + (size_t)l0 + (size_t)(lane & 15);
  const size_t row2 = 2 * (size_t)L;            // rows m+2 / m+3
  const size_t step = 4 * (size_t)L;

  v2f ones; ones.x = 1.0f; ones.y = 1.0f;       // A = 16x4 of ones
  v8f cs = {};                                  // accumulates sum(x)
  v8f cq = {};                                  // accumulates sum(x^2)

  for (int m = 0; m < M; m += 4) {
    float e0 = px[0];                           // rows m   / m+1
    float e1 = px[row2];                        // rows m+2 / m+3
    v2f bv; bv.x = e0;      bv.y = e1;
    v2f bq; bq.x = e0 * e0; bq.y = e1 * e1;
    cs = __builtin_amdgcn_wmma_f32_16x16x4_f32(
        false, ones, false, bv, (short)0, cs, false, false);
    cq = __builtin_amdgcn_wmma_f32_16x16x4_f32(
        false, ones, false, bq, (short)0, cq, false, false);
    px += step;
  }

  // Row M=0 of D lives in accumulator VGPR 0, lanes 0..15, N = lane.
  if (lane < 16) {
    s [(size_t)b * L + l0 + lane] = cs[0];
    s2[(size_t)b * L + l0 + lane] = cq[0];
  }
}

// ---------------------------------------------------------------------------
// Phase 2: per-batch inclusive prefix scan over L of (s, s2), converted in
// place into (mean, rstd). 8 blocks x 256 threads; 63 chunks of 256 with a
// block-wide Hillis-Steele scan in LDS and a running carry. ~2 MB of traffic,
// negligible versus the streaming passes.
// ---------------------------------------------------------------------------
__global__ void __launch_bounds__(256) cln_scan(
    float* __restrict__ s, float* __restrict__ s2, int M, int L)
{
  const int b   = (int)blockIdx.x;
  const int tid = (int)threadIdx.x;
  __shared__ float sh[256];
  __shared__ float sq[256];

  float c1 = 0.0f, c2 = 0.0f;                   // carry across chunks
  for (int base = 0; base < L; base += 256) {
    const int i = base + tid;
    float v = (i < L) ? s [(size_t)b * L + i] : 0.0f;
    float w = (i < L) ? s2[(size_t)b * L + i] : 0.0f;
    sh[tid] = v; sq[tid] = w;
    __syncthreads();
    for (int off = 1; off < 256; off <<= 1) {
      float av = (tid >= off) ? sh[tid - off] : 0.0f;
      float aw = (tid >= off) ? sq[tid - off] : 0.0f;
      __syncthreads();
      sh[tid] += av; sq[tid] += aw;
      __syncthreads();
    }
    const float inc1 = sh[tid] + c1;
    const float inc2 = sq[tid] + c2;
    if (i < L) {
      const float cnt  = (float)M * (float)(i + 1);
      const float mean = inc1 / cnt;
      const float var  = inc2 / cnt - mean * mean;
      s [(size_t)b * L + i] = mean;                     // mean in place
      s2[(size_t)b * L + i] = rsqrtf(var + CLN_EPS);    // rstd in place
    }
    c1 += sh[255];                                      // chunk total
    c2 += sq[255];
    __syncthreads();                                    // before LDS reuse
  }
}

// ---------------------------------------------------------------------------
// Phase 3: streaming apply, float4 (b128) loads/stores along L.
// out = gamma[b,m] * (x - mean[b,l]) * rstd[b,l] + beta[b,m]
// ---------------------------------------------------------------------------
__global__ void __launch_bounds__(256) cln_apply(
    const float* __restrict__ x, const float* __restrict__ gamma,
    const float* __restrict__ beta, const float* __restrict__ mean,
    const float* __restrict__ rstd, float* __restrict__ out,
    int B, int M, int L)
{
  const int    l4   = L >> 2;
  const size_t nvec = (size_t)B * (size_t)M * (size_t)l4;
  const size_t j    = (size_t)blockIdx.x * blockDim.x + threadIdx.x;
  if (j >= nvec) return;

  const int    lv = (int)(j % (size_t)l4);
  const size_t bm = j / (size_t)l4;
  const int    b  = (int)(bm / (size_t)M);

  const float4 xv = reinterpret_cast<const float4*>(x)[j];
  const float4 mn = reinterpret_cast<const float4*>(mean)[(size_t)b * l4 + lv];
  const float4 rs = reinterpret_cast<const float4*>(rstd)[(size_t)b * l4 + lv];
  const float  g  = gamma[bm];
  const float  bt = beta[bm];

  float4 o;
  o.x = g * (xv.x - mn.x) * rs.x + bt;
  o.y = g * (xv.y - mn.y) * rs.y + bt;
  o.z = g * (xv.z - mn.z) * rs.z + bt;
  o.w = g * (xv.w - mn.w) * rs.w + bt;
  reinterpret_cast<float4*>(out)[j] = o;
}

// ---------------------------------------------------------------------------
extern "C" void kernel_launch(void* const* d_in, const int* in_sizes, int n_in,
                              void* d_out, int out_size, void* d_ws, size_t ws_size,
                              hipStream_t stream) {
  const float* x     = (const float*)d_in[0];
  const float* gamma = (const float*)d_in[1];
  const float* beta  = (const float*)d_in[2];
  float*       out   = (float*)d_out;

  // Shapes: x=[B,M,L], gamma/beta=[B,M,1]. B fixed at 8 per reference.
  const int B  = 8;
  const int BM = in_sizes[1];                 // B*M
  const int M  = BM / B;                      // 512
  const int L  = in_sizes[0] / BM;            // 16000

  float* s  = (float*)d_ws;                   // [B, L] -> becomes mean
  float* s2 = s + (size_t)B * (size_t)L;      // [B, L] -> becomes rstd

  // Phase 1: one wave32 per (b, 16-l) tile.
  const int waves   = B * (L >> 4);           // 8000
  const int threads = 256;                    // 8 waves/block
  const int blocks1 = (waves * 32 + threads - 1) / threads;
  cln_colsum_wmma<<<blocks1, threads, 0, stream>>>(x, s, s2, B, M, L);

  // Phase 2: one block per batch row.
  cln_scan<<<B, 256, 0, stream>>>(s, s2, M, L);

  // Phase 3: vectorized elementwise apply.
  const size_t nvec   = (size_t)B * (size_t)M * (size_t)(L >> 2);
  const int    blocks3 = (int)((nvec + 255) / 256);
  cln_apply<<<blocks3, 256, 0, stream>>>(x, gamma, beta, s, s2, out, B, M, L);
}